// RK4Simulator_21663815041361
// MI455X (gfx1250) — compile-verified
//
#include <hip/hip_runtime.h>
#include <hip/hip_bf16.h>

typedef __attribute__((ext_vector_type(2))) float v2f;
typedef __attribute__((ext_vector_type(8))) float v8f;

#define HID 64
#define NTILE 4  // HID / 16

static __device__ __forceinline__ float fast_tanh(float x) {
#if __has_builtin(__builtin_amdgcn_tanhf)
  return __builtin_amdgcn_tanhf(x);
#else
  float r;
  asm volatile("v_tanh_f32 %0, %1" : "=v"(r) : "v"(x));
  return r;
#endif
}

// Exchange value with the opposite 16-lane half (lane L <-> L^16), VALU-only.
static __device__ __forceinline__ float xhalf(float v) {
#if __has_builtin(__builtin_amdgcn_permlanex16)
  int i = __builtin_bit_cast(int, v);
  i = __builtin_amdgcn_permlanex16(i, i, 0x76543210, 0xfedcba98, false, false);
  return __builtin_bit_cast(float, i);
#else
  return __shfl_xor(v, 16, 32);
#endif
}

struct LaneWeights {
  v2f A[NTILE];      // W1^T tiles, bias b1 folded into K=3 row (B K=3 carries 1.0)
  v2f w2[NTILE][8];  // W2 row pairs (out0,out1) for this lane's hidden slots
  v2f b2v;
};

// f(x,u) for 16 trajectories per wave.
// D-tile layout: lane (mod 16) = trajectory, VGPR r = hidden t*16 + (lane<16?0:8) + r.
// Each trajectory's 64 hidden values live in lanes L and L+16 -> one permlane swap.
static __device__ __forceinline__ v2f nn_f(float ex0, float ex1, float uval,
                                           bool lower, const LaneWeights& W) {
  v2f B;
  B[0] = lower ? ex0 : uval;  // K0 (x0) | K2 (u)
  B[1] = lower ? ex1 : 1.0f;  // K1 (x1) | K3 (bias one)
  v2f acc = {0.0f, 0.0f};
#pragma unroll
  for (int t = 0; t < NTILE; ++t) {
    v8f c = {};
    v8f d = __builtin_amdgcn_wmma_f32_16x16x4_f32(false, W.A[t], false, B,
                                                  (short)0, c, false, false);
#pragma unroll
    for (int r = 0; r < 8; ++r) {
      float hv = fast_tanh(d[r]);
      v2f hvv;
      hvv[0] = hv;
      hvv[1] = hv;
      // both output dims in one packed FMA (v_pk_fma_f32)
      acc = __builtin_elementwise_fma(hvv, W.w2[t][r], acc);
    }
  }
  v2f sw;
  sw[0] = xhalf(acc[0]);
  sw[1] = xhalf(acc[1]);
  return acc + sw + W.b2v;  // v_pk_add_f32
}

__global__ __launch_bounds__(256, 1) void rk4_sim_kernel(
    const float* __restrict__ x0, const float* __restrict__ u,
    const float* __restrict__ W1, const float* __restrict__ b1,
    const float* __restrict__ W2, const float* __restrict__ b2,
    float* __restrict__ out, int m, int q) {
  const int lane = (int)(threadIdx.x & 31u);
  const int ln = lane & 15;
  const bool lower = lane < 16;
  const int wave =
      (int)blockIdx.x * ((int)blockDim.x >> 5) + ((int)threadIdx.x >> 5);
  const int trajBase = wave * 16;
  int traj = trajBase + ln;
  const bool valid = traj < q;
  if (traj >= q) traj = q - 1;

  // ---- one-time per-lane weight staging (loop-invariant VGPRs) ----
  LaneWeights W;
#pragma unroll
  for (int t = 0; t < NTILE; ++t) {
    const int col = t * 16 + ln;
    // A-matrix 16x4 f32 layout: lanes 0-15 -> K0/K1, lanes 16-31 -> K2/K3
    W.A[t][0] = lower ? W1[0 * HID + col] : W1[2 * HID + col];
    W.A[t][1] = lower ? W1[1 * HID + col] : b1[col];
    const int hbase = t * 16 + (lower ? 0 : 8);
#pragma unroll
    for (int r = 0; r < 8; ++r)
      W.w2[t][r] = *(const v2f*)(W2 + (size_t)(hbase + r) * 2);
  }
  W.b2v[0] = b2[0];
  W.b2v[1] = b2[1];

  // Pin all weights in VGPRs: the empty asm's result is not rematerializable,
  // so the scheduler cannot re-stream the invariant loads inside the loop.
#pragma unroll
  for (int t = 0; t < NTILE; ++t) {
    asm volatile("" : "+v"(W.A[t]));
#pragma unroll
    for (int r = 0; r < 8; ++r) asm volatile("" : "+v"(W.w2[t][r]));
  }
  asm volatile("" : "+v"(W.b2v));

  float xs0 = x0[(size_t)traj * 2 + 0];
  float xs1 = x0[(size_t)traj * 2 + 1];

  // 32-bit byte-offset addressing (u = 64MB, out = 128MB: fits u32) so the
  // backend can use saddr + 32-bit voffset and one v_add_nc_u32 per stream.
  const char* ub = (const char*)u;
  char* ob = (char*)out;
  const uint32_t ustride = (uint32_t)q * 4u;
  const uint32_t ostride = (uint32_t)q * 8u;
  uint32_t uoff = (uint32_t)traj * 4u + ustride;  // -> u[step+1][traj]
  uint32_t ooff = (uint32_t)traj * 8u;            // -> out[step][traj][0]

  float ucur =
      __builtin_nontemporal_load((const float*)(ub + (uint32_t)traj * 4u));

  const float dt2 = 0.5f;           // TS / 2
  const float sixth = 1.0f / 6.0f;  // TS / 6

  for (int step = 0; step < m; ++step) {
    // double-buffer next u; prefetch a few steps ahead (global_prefetch_b8)
    float unxt = 0.0f;
    if (step + 1 < m)
      unxt = __builtin_nontemporal_load((const float*)(ub + uoff));
    if (step + 8 < m)
      __builtin_prefetch((const float*)(ub + uoff + 7u * ustride), 0, 0);

    // output is state BEFORE update (matches lax.scan)
    if (lower && valid) {
      v2f xv;
      xv[0] = xs0;
      xv[1] = xs1;
      __builtin_nontemporal_store(xv, (v2f*)(ob + ooff));
    }

    const v2f k1 = nn_f(xs0, xs1, ucur, lower, W);
    const v2f k2 = nn_f(__builtin_fmaf(dt2, k1[0], xs0),
                        __builtin_fmaf(dt2, k1[1], xs1), ucur, lower, W);
    const v2f k3 = nn_f(__builtin_fmaf(dt2, k2[0], xs0),
                        __builtin_fmaf(dt2, k2[1], xs1), ucur, lower, W);
    const v2f k4 = nn_f(xs0 + k3[0], xs1 + k3[1], ucur, lower, W);

    xs0 += sixth * (k1[0] + 2.0f * k2[0] + 2.0f * k3[0] + k4[0]);
    xs1 += sixth * (k1[1] + 2.0f * k2[1] + 2.0f * k3[1] + k4[1]);

    ucur = unxt;
    uoff += ustride;
    ooff += ostride;
  }
}

extern "C" void kernel_launch(void* const* d_in, const int* in_sizes, int n_in,
                              void* d_out, int out_size, void* d_ws,
                              size_t ws_size, hipStream_t stream) {
  const float* x0 = (const float*)d_in[0];
  const float* u = (const float*)d_in[1];
  const float* W1 = (const float*)d_in[2];
  const float* b1 = (const float*)d_in[3];
  const float* W2 = (const float*)d_in[4];
  const float* b2 = (const float*)d_in[5];
  float* out = (float*)d_out;

  const int q = in_sizes[0] / 2;                // x0 is (q, 2)
  const int m = (q > 0) ? in_sizes[1] / q : 0;  // u is (m, q, 1)

  const int wavesNeeded = (q + 15) / 16;  // 16 trajectories per wave32
  const int wavesPerBlock = 8;            // 256 threads
  int blocks = (wavesNeeded + wavesPerBlock - 1) / wavesPerBlock;
  if (blocks < 1) blocks = 1;

  rk4_sim_kernel<<<blocks, 256, 0, stream>>>(x0, u, W1, b1, W2, b2, out, m, q);
}